// KCanyon3D_73607149519208
// MI455X (gfx1250) — compile-verified
//
#include <hip/hip_runtime.h>
#include <math.h>

// ---------------------------------------------------------------------------
// velocity = -grad(U) via central finite differences, H = 1e-4
//   U(p) = 0.5*a*r^2 + (1-w(theta)) * 0.5*b*r^2*theta^2
//   theta = acos(clamp(p_hat . dir, -1, 1)), w = smoothstep over [pi/2±pi/8]
//
// MI455X reasoning: 24 B/point streamed once -> 201 MB @ 23.3 TB/s = 8.6 us
// floor; kernel is VALU-bound (~170 instr/point). Round-3 showed the gfx1250
// backend does NOT form packed fp32 VOP3P ops (v2f32 vectorization regressed
// +150 instr), so this is the scalar-optimal form: v_rsq_f32 radius path,
// 11-instr minimax acos (fabs -> src modifier, clamps -> v_med3), incremental
// r^2/dot updates across the 6 FD evals, and (1-w) folded into one fma.
// Memory path: 128-bit NT loads/stores (single-pass stream), 4 points per
// wave32 lane, 256-thread (8-wave) blocks.
// ---------------------------------------------------------------------------

typedef __attribute__((ext_vector_type(4))) float fvec4;

#define FD_H    1.0e-4f
#define INV_2H  5000.0f                    // 1/(2H)
#define INV_W   1.2732395447351628f        // 4/pi = 1/(HIGH-LOW)
#define PI_F    3.14159265358979323846f

// A&S 4.4.45 degree-7 minimax: acos(x) = sqrt(1-x)*p(x), x in [0,1], |eps|<=2e-8.
__device__ __forceinline__ float acos_fast(float x) {
    float ax = __builtin_fabsf(x);                 // folds into fma src modifier
    float p  = fmaf(ax, -0.0012624911f, 0.0066700901f);
    p = fmaf(ax, p, -0.0170881256f);
    p = fmaf(ax, p,  0.0308918810f);
    p = fmaf(ax, p, -0.0501743046f);
    p = fmaf(ax, p,  0.0889789874f);
    p = fmaf(ax, p, -0.2145988016f);
    p = fmaf(ax, p,  1.5707963050f);
    float s = __builtin_amdgcn_sqrtf(1.0f - ax);   // raw v_sqrt_f32 (~1 ulp)
    float t = s * p;
    return (x < 0.0f) ? (PI_F - t) : t;
}

// One potential evaluation from precomputed r^2 and un-normalized dot.
__device__ __forceinline__ float pot_eval(float r2, float dotraw,
                                          float ha, float hb) {
    float inv_r = __builtin_amdgcn_rsqf(r2);       // single v_rsq_f32
    float dot   = dotraw * inv_r;
    dot = fminf(fmaxf(dot, -1.0f), 1.0f);          // v_med3_num_f32, NaN-safe
    float th = acos_fast(dot);
    // smoothstep arg: s = clamp((th-LOW)/(pi/4)) ; LOW/(pi/4) == 1.5 exactly
    float s  = fminf(fmaxf(fmaf(th, INV_W, -1.5f), 0.0f), 1.0f);
    // 1 - w = 1 - s^2*(3-2s) folded into a single fma
    float f   = fmaf(-2.0f, s, 3.0f);
    float omw = fmaf(-(s * s), f, 1.0f);
    float term = (omw * (hb * r2)) * (th * th);
    float pot  = fmaf(ha, r2, term);
    return (r2 == 0.0f) ? 0.0f : pot;
}

__device__ __forceinline__ void canyon_grad(float x, float y, float z,
                                            float ha, float hb,
                                            float dx, float dy, float dz,
                                            float& vx, float& vy, float& vz) {
    float r2b = fmaf(x, x, fmaf(y, y, z * z));
    float db  = fmaf(x, dx, fmaf(y, dy, z * dz));
    float up, um, r2p, r2m;

    // axis x: r2(x±H) = r2b ± H*(2x±H) ; dot(x±H) = db ± H*dx
    r2p = fmaf( FD_H, fmaf(2.0f, x,  FD_H), r2b);
    r2m = fmaf(-FD_H, fmaf(2.0f, x, -FD_H), r2b);
    up  = pot_eval(r2p, fmaf( FD_H, dx, db), ha, hb);
    um  = pot_eval(r2m, fmaf(-FD_H, dx, db), ha, hb);
    vx  = (um - up) * INV_2H;                      // -(up-um)/(2H)

    // axis y
    r2p = fmaf( FD_H, fmaf(2.0f, y,  FD_H), r2b);
    r2m = fmaf(-FD_H, fmaf(2.0f, y, -FD_H), r2b);
    up  = pot_eval(r2p, fmaf( FD_H, dy, db), ha, hb);
    um  = pot_eval(r2m, fmaf(-FD_H, dy, db), ha, hb);
    vy  = (um - up) * INV_2H;

    // axis z
    r2p = fmaf( FD_H, fmaf(2.0f, z,  FD_H), r2b);
    r2m = fmaf(-FD_H, fmaf(2.0f, z, -FD_H), r2b);
    up  = pot_eval(r2p, fmaf( FD_H, dz, db), ha, hb);
    um  = pot_eval(r2m, fmaf(-FD_H, dz, db), ha, hb);
    vz  = (um - up) * INV_2H;
}

// Main path: 4 points per lane -> 3x b128 NT loads, 3x b128 NT stores.
__global__ __launch_bounds__(256) void
KCanyon3D_vel4(const fvec4* __restrict__ in4, fvec4* __restrict__ out4,
               const float* __restrict__ a_p, const float* __restrict__ b_p,
               const float* __restrict__ dir_p, int nquad) {
    int t = blockIdx.x * blockDim.x + threadIdx.x;
    if (t >= nquad) return;

    // Uniform params -> scalar loads, broadcast to the wave.
    float ha = 0.5f * fminf(fmaxf(a_p[0], 0.0f), 20.0f);
    float hb = 0.5f * fminf(fmaxf(b_p[0], 0.0f), 20.0f);
    float dx = dir_p[0], dy = dir_p[1], dz = dir_p[2];

    long base = 3l * t;
    fvec4 p0 = __builtin_nontemporal_load(in4 + base + 0);
    fvec4 p1 = __builtin_nontemporal_load(in4 + base + 1);
    fvec4 p2 = __builtin_nontemporal_load(in4 + base + 2);

    float o0, o1, o2, o3, o4, o5, o6, o7, o8, o9, o10, o11;
    canyon_grad(p0.x, p0.y, p0.z, ha, hb, dx, dy, dz, o0,  o1,  o2);
    canyon_grad(p0.w, p1.x, p1.y, ha, hb, dx, dy, dz, o3,  o4,  o5);
    canyon_grad(p1.z, p1.w, p2.x, ha, hb, dx, dy, dz, o6,  o7,  o8);
    canyon_grad(p2.y, p2.z, p2.w, ha, hb, dx, dy, dz, o9,  o10, o11);

    fvec4 q0 = {o0, o1, o2,  o3};
    fvec4 q1 = {o4, o5, o6,  o7};
    fvec4 q2 = {o8, o9, o10, o11};
    __builtin_nontemporal_store(q0, out4 + base + 0);
    __builtin_nontemporal_store(q1, out4 + base + 1);
    __builtin_nontemporal_store(q2, out4 + base + 2);
}

// Tail path: one point per lane (only used if B % 4 != 0).
__global__ __launch_bounds__(256) void
KCanyon3D_vel1(const float* __restrict__ in, float* __restrict__ out,
               const float* __restrict__ a_p, const float* __restrict__ b_p,
               const float* __restrict__ dir_p, int start, int npts) {
    int i = start + blockIdx.x * blockDim.x + threadIdx.x;
    if (i >= npts) return;
    float ha = 0.5f * fminf(fmaxf(a_p[0], 0.0f), 20.0f);
    float hb = 0.5f * fminf(fmaxf(b_p[0], 0.0f), 20.0f);
    float dx = dir_p[0], dy = dir_p[1], dz = dir_p[2];
    float x = in[3l * i + 0], y = in[3l * i + 1], z = in[3l * i + 2];
    float vx, vy, vz;
    canyon_grad(x, y, z, ha, hb, dx, dy, dz, vx, vy, vz);
    out[3l * i + 0] = vx;
    out[3l * i + 1] = vy;
    out[3l * i + 2] = vz;
}

extern "C" void kernel_launch(void* const* d_in, const int* in_sizes, int n_in,
                              void* d_out, int out_size, void* d_ws, size_t ws_size,
                              hipStream_t stream) {
    const float* xyz = (const float*)d_in[0];
    const float* a_p = (const float*)d_in[1];
    const float* b_p = (const float*)d_in[2];
    const float* dir = (const float*)d_in[3];
    float* out = (float*)d_out;

    const int B     = in_sizes[0] / 3;   // number of points
    const int nquad = B / 4;             // 4-point groups (12 floats = 3 x float4)
    const int rem   = B - 4 * nquad;

    const int threads = 256;             // 8 wave32 waves per block
    if (nquad > 0) {
        int blocks = (nquad + threads - 1) / threads;
        KCanyon3D_vel4<<<blocks, threads, 0, stream>>>(
            (const fvec4*)xyz, (fvec4*)out, a_p, b_p, dir, nquad);
    }
    if (rem > 0) {
        int blocks = (rem + threads - 1) / threads;
        KCanyon3D_vel1<<<blocks, threads, 0, stream>>>(
            xyz, out, a_p, b_p, dir, 4 * nquad, B);
    }
}